// Crossalign_k_60421599920362
// MI455X (gfx1250) — compile-verified
//
#include <hip/hip_runtime.h>
#include <math.h>

// Problem constants (from the reference)
#define NB   2
#define NPT  8192
#define IC   256
#define PC   128
#define QKV  128
#define IMH  120
#define IMW  160
#define HW   (IMH * IMW)   // 19200
#define KNN  4
#define CHUNK 2048

typedef __attribute__((ext_vector_type(2))) float v2f;
typedef __attribute__((ext_vector_type(8))) float v8f;

static __device__ __forceinline__ v8f wmma_f32(v2f a, v2f b, v8f c) {
  // D = A(16x4) * B(4x16) + C(16x16), full f32
  return __builtin_amdgcn_wmma_f32_16x16x4_f32(false, a, false, b, (short)0, c,
                                               false, false);
}

// ---------------------------------------------------------------------------
// Kernel 1: transpose image [B, IC, HW] -> imgT [B, HW, IC] (coalesced tiles)
// ---------------------------------------------------------------------------
__global__ void k_transpose(const float* __restrict__ img, float* __restrict__ imgT) {
  __shared__ float tile[32][33];
  const int b   = blockIdx.z;
  const int hw0 = blockIdx.x * 32;
  const int c0  = blockIdx.y * 32;
  const int tx  = threadIdx.x;   // 0..31
  const int ty  = threadIdx.y;   // 0..7
#pragma unroll
  for (int i = 0; i < 4; ++i) {
    const int c = c0 + ty + i * 8;
    tile[ty + i * 8][tx] = img[((size_t)b * IC + c) * HW + hw0 + tx];
  }
  __syncthreads();
#pragma unroll
  for (int i = 0; i < 4; ++i) {
    const int hw = hw0 + ty + i * 8;
    imgT[((size_t)b * HW + hw) * IC + c0 + tx] = tile[tx][ty + i * 8];
  }
}

// ---------------------------------------------------------------------------
// Kernel 2: brute-force KNN (top-4 incl. self) over li_xyz [B, N, 3]
// One thread per query point; candidate xyz staged in LDS chunks.
// ---------------------------------------------------------------------------
__global__ void __launch_bounds__(128) k_knn(const float* __restrict__ xyz,
                                             int* __restrict__ knn_idx) {
  const int b = blockIdx.y;
  const int q = blockIdx.x * 128 + threadIdx.x;
  const float* X = xyz + (size_t)b * NPT * 3;

  __shared__ float sx[CHUNK], sy[CHUNK], sz[CHUNK];

  const float qx = X[q * 3 + 0];
  const float qy = X[q * 3 + 1];
  const float qz = X[q * 3 + 2];

  float d0 = 3.0e38f, d1 = 3.0e38f, d2 = 3.0e38f, d3 = 3.0e38f;
  int   i0 = 0,        i1 = 0,       i2 = 0,       i3 = 0;

  for (int j0 = 0; j0 < NPT; j0 += CHUNK) {
    __syncthreads();
    for (int j = threadIdx.x; j < CHUNK; j += 128) {
      sx[j] = X[(j0 + j) * 3 + 0];
      sy[j] = X[(j0 + j) * 3 + 1];
      sz[j] = X[(j0 + j) * 3 + 2];
    }
    __syncthreads();
    for (int j = 0; j < CHUNK; ++j) {
      const float dx = sx[j] - qx;
      const float dy = sy[j] - qy;
      const float dz = sz[j] - qz;
      const float d  = dx * dx + dy * dy + dz * dz;
      if (d < d3) {                      // wave mostly skips this after warm-up
        const int idx = j0 + j;
        if (d < d2) {
          d3 = d2; i3 = i2;
          if (d < d1) {
            d2 = d1; i2 = i1;
            if (d < d0) { d1 = d0; i1 = i0; d0 = d; i0 = idx; }
            else        { d1 = d;  i1 = idx; }
          } else { d2 = d; i2 = idx; }
        } else { d3 = d; i3 = idx; }
      }
    }
  }
  *(int4*)&knn_idx[((size_t)b * NPT + q) * KNN] = make_int4(i0, i1, i2, i3);
}

// ---------------------------------------------------------------------------
// Kernel 3: per-point projections with f32 WMMA.
//   Q = pts@W_q+b_q, K = img_pt@W_k+b_k, V = img_pt@W_v+b_v
// Emits dot[b,m] = (Q.K)/sqrt(128) and V_feat[b,m,128]. (Q,K never stored.)
// 1 block = 16 points, 4 waves; wave w owns output columns [32w, 32w+32).
// ---------------------------------------------------------------------------
__global__ void __launch_bounds__(128) k_proj(
    const float* __restrict__ point_feas, const float* __restrict__ imgT,
    const int* __restrict__ li_index,
    const float* __restrict__ W_q, const float* __restrict__ b_q,
    const float* __restrict__ W_k, const float* __restrict__ b_k,
    const float* __restrict__ W_v, const float* __restrict__ b_v,
    float* __restrict__ dot_out, float* __restrict__ V_out) {
  __shared__ float sP[16][PC];   // point features, 16 rows x 128
  __shared__ float sI[16][IC];   // gathered image features, 16 rows x 256
  __shared__ float sdot[16];

  const int tile = blockIdx.x;            // B*N/16 tiles
  const int b    = tile / (NPT / 16);
  const int m0   = (tile % (NPT / 16)) * 16;
  const int tid  = threadIdx.x;

  for (int i = tid; i < 16 * PC; i += 128) {
    const int m = i % 16, c = i / 16;
    sP[m][c] = point_feas[((size_t)b * PC + c) * NPT + m0 + m];
  }
  for (int i = tid; i < 16 * IC; i += 128) {    // contiguous 1KB rows of imgT
    const int m = i / IC, c = i % IC;
    const int row = li_index[b * NPT + m0 + m];
    sI[m][c] = imgT[((size_t)b * HW + row) * IC + c];
  }
  if (tid < 16) sdot[tid] = 0.0f;
  __syncthreads();

  const int wave = tid / 32, lane = tid % 32;
  const int lm = lane % 16, kh = lane / 16;   // A: K-half; C: M-half

  v8f accq[2], acck[2], accv[2];
#pragma unroll
  for (int nt = 0; nt < 2; ++nt) {
    const int col = wave * 32 + nt * 16 + lm;
    const float bq = b_q[col], bk = b_k[col], bv = b_v[col];
#pragma unroll
    for (int r = 0; r < 8; ++r) { accq[nt][r] = bq; acck[nt][r] = bk; accv[nt][r] = bv; }
  }

  // Q projection: inner dim PC=128
  for (int kk = 0; kk < PC; kk += 4) {
    const int k0 = kk + 2 * kh;
    v2f a; a.x = sP[lm][k0]; a.y = sP[lm][k0 + 1];
#pragma unroll
    for (int nt = 0; nt < 2; ++nt) {
      const int col = wave * 32 + nt * 16 + lm;
      v2f bb; bb.x = W_q[k0 * QKV + col]; bb.y = W_q[(k0 + 1) * QKV + col];
      accq[nt] = wmma_f32(a, bb, accq[nt]);
    }
  }
  // K and V projections: inner dim IC=256
  for (int kk = 0; kk < IC; kk += 4) {
    const int k0 = kk + 2 * kh;
    v2f a; a.x = sI[lm][k0]; a.y = sI[lm][k0 + 1];
#pragma unroll
    for (int nt = 0; nt < 2; ++nt) {
      const int col = wave * 32 + nt * 16 + lm;
      v2f bk2; bk2.x = W_k[k0 * QKV + col]; bk2.y = W_k[(k0 + 1) * QKV + col];
      acck[nt] = wmma_f32(a, bk2, acck[nt]);
      v2f bv2; bv2.x = W_v[k0 * QKV + col]; bv2.y = W_v[(k0 + 1) * QKV + col];
      accv[nt] = wmma_f32(a, bv2, accv[nt]);
    }
  }

  // dot[m] = sum_col Q*K : per-lane partials over its 8 rows, LDS f32 atomics
  float p[8] = {0, 0, 0, 0, 0, 0, 0, 0};
#pragma unroll
  for (int nt = 0; nt < 2; ++nt)
#pragma unroll
    for (int r = 0; r < 8; ++r) p[r] += accq[nt][r] * acck[nt][r];
#pragma unroll
  for (int r = 0; r < 8; ++r) atomicAdd(&sdot[r + 8 * kh], p[r]);

  // V_feat stores (row m = r + 8*kh, column col)
#pragma unroll
  for (int nt = 0; nt < 2; ++nt) {
    const int col = wave * 32 + nt * 16 + lm;
#pragma unroll
    for (int r = 0; r < 8; ++r) {
      const int m = r + 8 * kh;
      V_out[((size_t)(b * NPT + m0 + m)) * QKV + col] = accv[nt][r];
    }
  }
  __syncthreads();
  if (tid < 16)
    dot_out[b * NPT + m0 + tid] = sdot[tid] * 0.08838834764831845f; // 1/sqrt(128)
}

// ---------------------------------------------------------------------------
// Kernel 4: softmax over 4 gathered dots + weighted V sum (incl. /K for mean)
// 1 block per point, thread c handles channel c of vbar.
// ---------------------------------------------------------------------------
__global__ void __launch_bounds__(128) k_attn(const int* __restrict__ knn_idx,
                                              const float* __restrict__ dot,
                                              const float* __restrict__ V,
                                              float* __restrict__ vbar) {
  const int bn = blockIdx.x;          // 0 .. B*N-1
  const int b  = bn / NPT;
  const int c  = threadIdx.x;
  const int4 id = *(const int4*)&knn_idx[(size_t)bn * KNN];
  const int base = b * NPT;
  const float a0 = dot[base + id.x], a1 = dot[base + id.y];
  const float a2 = dot[base + id.z], a3 = dot[base + id.w];
  const float mx = fmaxf(fmaxf(a0, a1), fmaxf(a2, a3));
  const float e0 = expf(a0 - mx), e1 = expf(a1 - mx);
  const float e2 = expf(a2 - mx), e3 = expf(a3 - mx);
  const float inv = 0.25f / (e0 + e1 + e2 + e3);   // softmax AND mean over K=4
  vbar[(size_t)bn * QKV + c] =
      e0 * inv * V[(size_t)(base + id.x) * QKV + c] +
      e1 * inv * V[(size_t)(base + id.y) * QKV + c] +
      e2 * inv * V[(size_t)(base + id.z) * QKV + c] +
      e3 * inv * V[(size_t)(base + id.w) * QKV + c];
}

// ---------------------------------------------------------------------------
// Kernel 5: out[b,c,n] = vbar @ W_o + b_o, WMMA f32, transposed vectorized store
// 1 block = 16 points, 4 waves; wave w owns 64 of the 256 output channels.
// ---------------------------------------------------------------------------
__global__ void __launch_bounds__(128) k_out(const float* __restrict__ vbar,
                                             const float* __restrict__ W_o,
                                             const float* __restrict__ b_o,
                                             float* __restrict__ out) {
  __shared__ float sA[16][QKV];
  const int tile = blockIdx.x;
  const int b    = tile / (NPT / 16);
  const int m0   = (tile % (NPT / 16)) * 16;
  const int tid  = threadIdx.x;

  for (int i = tid; i < 16 * QKV; i += 128) {
    const int m = i / QKV, c = i % QKV;
    sA[m][c] = vbar[((size_t)(b * NPT + m0 + m)) * QKV + c];
  }
  __syncthreads();

  const int wave = tid / 32, lane = tid % 32;
  const int lm = lane % 16, kh = lane / 16;

  v8f acc[4];
#pragma unroll
  for (int nt = 0; nt < 4; ++nt) {
    const float bo = b_o[wave * 64 + nt * 16 + lm];
#pragma unroll
    for (int r = 0; r < 8; ++r) acc[nt][r] = bo;
  }

  for (int kk = 0; kk < QKV; kk += 4) {
    const int k0 = kk + 2 * kh;
    v2f a; a.x = sA[lm][k0]; a.y = sA[lm][k0 + 1];
#pragma unroll
    for (int nt = 0; nt < 4; ++nt) {
      const int col = wave * 64 + nt * 16 + lm;
      v2f bb; bb.x = W_o[k0 * IC + col]; bb.y = W_o[(k0 + 1) * IC + col];
      acc[nt] = wmma_f32(a, bb, acc[nt]);
    }
  }

  // out[b][col][m0 + 8*kh + r]: 8 consecutive floats per lane -> 2x float4
#pragma unroll
  for (int nt = 0; nt < 4; ++nt) {
    const int col = wave * 64 + nt * 16 + lm;
    const size_t off = ((size_t)b * IC + col) * NPT + m0 + 8 * kh;
    *(float4*)&out[off]     = make_float4(acc[nt][0], acc[nt][1], acc[nt][2], acc[nt][3]);
    *(float4*)&out[off + 4] = make_float4(acc[nt][4], acc[nt][5], acc[nt][6], acc[nt][7]);
  }
}

// ---------------------------------------------------------------------------
extern "C" void kernel_launch(void* const* d_in, const int* in_sizes, int n_in,
                              void* d_out, int out_size, void* d_ws, size_t ws_size,
                              hipStream_t stream) {
  (void)in_sizes; (void)n_in; (void)out_size; (void)ws_size;

  const float* img      = (const float*)d_in[0];   // [B, IC, H, W]
  const float* pfe      = (const float*)d_in[1];   // [B, PC, N]
  const float* xyz      = (const float*)d_in[2];   // [B, N, 3]
  const float* W_q      = (const float*)d_in[3];
  const float* b_q      = (const float*)d_in[4];
  const float* W_k      = (const float*)d_in[5];
  const float* b_k      = (const float*)d_in[6];
  const float* W_v      = (const float*)d_in[7];
  const float* b_v      = (const float*)d_in[8];
  const float* W_o      = (const float*)d_in[9];
  const float* b_o      = (const float*)d_in[10];
  const int*   li_index = (const int*)d_in[11];    // [B, N]
  float* out = (float*)d_out;                      // [B, IC, N]

  // Workspace layout (all offsets 256B aligned)
  char* ws = (char*)d_ws;
  float* imgT   = (float*)ws;                 // B*HW*IC   = 39,321,600 B
  ws += (size_t)NB * HW * IC * sizeof(float);
  float* V_feat = (float*)ws;                 // B*N*QKV   =  8,388,608 B
  ws += (size_t)NB * NPT * QKV * sizeof(float);
  float* vbar   = (float*)ws;                 // B*N*QKV   =  8,388,608 B
  ws += (size_t)NB * NPT * QKV * sizeof(float);
  float* dotb   = (float*)ws;                 // B*N       =     65,536 B
  ws += (size_t)NB * NPT * sizeof(float);
  int*   knn    = (int*)ws;                   // B*N*4     =    262,144 B

  k_transpose<<<dim3(HW / 32, IC / 32, NB), dim3(32, 8), 0, stream>>>(img, imgT);
  k_knn<<<dim3(NPT / 128, NB), dim3(128), 0, stream>>>(xyz, knn);
  k_proj<<<dim3(NB * NPT / 16), dim3(128), 0, stream>>>(
      pfe, imgT, li_index, W_q, b_q, W_k, b_k, W_v, b_v, dotb, V_feat);
  k_attn<<<dim3(NB * NPT), dim3(128), 0, stream>>>(knn, dotb, V_feat, vbar);
  k_out<<<dim3(NB * NPT / 16), dim3(128), 0, stream>>>(vbar, W_o, b_o, out);
}